// TransfNet_30124900614320
// MI455X (gfx1250) — compile-verified
//
#include <hip/hip_runtime.h>
#include <math.h>

// ---- problem constants (match reference) ----
#define HEADS   8
#define HC      8
#define HID     64      // HEADS*HC == D_IN
#define OUTC    16
#define EPSF    1e-16f
#define INV_SQRT_HC 0.35355339059327373f  // 1/sqrt(8)

typedef float v2f __attribute__((ext_vector_type(2)));
typedef float v8f __attribute__((ext_vector_type(8)));

// ---------------- float atomic max (CAS trick, works for negatives) --------
__device__ __forceinline__ void atomicMaxF(float* addr, float val) {
    int* ai = (int*)addr;
    int old = *ai;
    while (val > __int_as_float(old)) {
        int assumed = old;
        old = atomicCAS(ai, assumed, __float_as_int(val));
        if (old == assumed) break;
    }
}

// ---------------- WMMA f32 GEMM with LDS staging ---------------------------
//   Y[M, NC] = X[M, 64] @ W[64, NC] + bias,  NC = COLT*16, K fixed at 64.
// 256 threads = 8 waves per block. Block stages W (64xNC) and its X row-block
// into LDS with coalesced b128 loads; each wave owns one 16x16 output tile and
// issues 16 fully-unrolled V_WMMA_F32_16X16X4_F32 ops (native f32 => exact
// vs the f32 reference). Fragment layouts per CDNA5 ISA:
//   A 16x4 f32: lanes 0-15 K={k0,k0+1}, lanes 16-31 K={k0+2,k0+3} (ds_load_b64)
//   C/D 16x16:  VGPR j -> M=j (+8 upper half-wave), N = lane&15
template<int COLT>
__global__ void __launch_bounds__(256)
gemm_lds_wmma(const float* __restrict__ X, const float* __restrict__ W,
              const float* __restrict__ bias, float* __restrict__ Y, int M) {
    constexpr int NC   = COLT * 16;   // output columns
    constexpr int ROWT = 8 / COLT;    // row tiles per block
    constexpr int RB   = ROWT * 16;   // rows per block
    constexpr int K    = 64;

    __shared__ float Ws[K * NC];
    __shared__ float Xs[RB * K];

    const int tid      = threadIdx.x;
    const int blockRow = blockIdx.x * RB;

    // stage W (64 x NC, contiguous) via float4
    {
        const float4* Wv  = (const float4*)W;
        float4*       Wsv = (float4*)Ws;
#pragma unroll
        for (int i = tid; i < (K * NC) / 4; i += 256) Wsv[i] = Wv[i];
    }
    // stage X row-block (RB x 64), row-clamped for the ragged tail
    {
        float4* Xsv = (float4*)Xs;
#pragma unroll
        for (int i = tid; i < RB * (K / 4); i += 256) {
            int r   = i >> 4;              // K/4 == 16 float4 per row
            int c   = i & 15;
            int row = blockRow + r;
            if (row >= M) row = M - 1;
            Xsv[i] = ((const float4*)(X + (long)row * K))[c];
        }
    }
    __syncthreads();

    const int wave = tid >> 5;
    const int lane = tid & 31;
    const int rt   = wave / COLT;
    const int ct   = wave % COLT;
    const int m    = lane & 15;
    const int hi   = lane >> 4;

    const float* xr = &Xs[(rt * 16 + m) * K];   // A-row for this lane
    const float* wc = &Ws[ct * 16 + m];         // B-column for this lane

    v8f acc = {0.f, 0.f, 0.f, 0.f, 0.f, 0.f, 0.f, 0.f};
#pragma unroll
    for (int k0 = 0; k0 < K; k0 += 4) {
        const int ka = k0 + (hi << 1);          // this half-wave's K pair
        v2f a = *(const v2f*)&xr[ka];           // ds_load_b64 (8B aligned)
        v2f b;
        b.x = wc[ka * NC];
        b.y = wc[(ka + 1) * NC];
        acc = __builtin_amdgcn_wmma_f32_16x16x4_f32(
                  /*neg_a=*/false, a, /*neg_b=*/false, b,
                  /*c_mod=*/(short)0, acc, /*reuse_a=*/false, /*reuse_b=*/false);
    }

    const int   n  = ct * 16 + m;
    const float bn = bias[n];
#pragma unroll
    for (int j = 0; j < 8; ++j) {
        const int r = blockRow + rt * 16 + j + (hi << 3);
        if (r < M) Y[(long)r * NC + n] = acc[j] + bn;
    }
}

// ---------------- fill buffer with a constant ------------------------------
__global__ void fill_kernel(float* __restrict__ p, float val, int n) {
    int i = blockIdx.x * blockDim.x + threadIdx.x;
    if (i < n) p[i] = val;
}

// ---------------- per-edge attention logits + segment max ------------------
// thread t -> (edge e = t/8, head h = t%8);  e_emb[h,c] = ew[e]*We[h*8+c]
__global__ void edge_alpha_kernel(const int* __restrict__ src, const int* __restrict__ dst,
                                  const float* __restrict__ ew, const float* __restrict__ We,
                                  const float* __restrict__ q, const float* __restrict__ k,
                                  float* __restrict__ alpha, float* __restrict__ amax,
                                  int E) {
    int t = blockIdx.x * blockDim.x + threadIdx.x;
    if (t >= E * HEADS) return;
    const int e = t >> 3, h = t & 7;
    const int s = src[e], d = dst[e];
    const float w = ew[e];
    const float4* qd = (const float4*)(q  + (long)d * HID + h * HC);  // 32B aligned
    const float4* ks = (const float4*)(k  + (long)s * HID + h * HC);
    const float4* we = (const float4*)(We + h * HC);
    float4 q0 = qd[0], q1 = qd[1];
    float4 k0 = ks[0], k1 = ks[1];
    float4 w0 = we[0], w1 = we[1];
    float acc = q0.x * (k0.x + w * w0.x) + q0.y * (k0.y + w * w0.y)
              + q0.z * (k0.z + w * w0.z) + q0.w * (k0.w + w * w0.w)
              + q1.x * (k1.x + w * w1.x) + q1.y * (k1.y + w * w1.y)
              + q1.z * (k1.z + w * w1.z) + q1.w * (k1.w + w * w1.w);
    acc *= INV_SQRT_HC;
    alpha[t] = acc;
    atomicMaxF(&amax[(long)d * HEADS + h], acc);
}

// ---------------- exp(alpha - amax[dst]) + segment sum ---------------------
__global__ void edge_exp_kernel(const int* __restrict__ dst,
                                const float* __restrict__ amax,
                                float* __restrict__ alpha, float* __restrict__ denom,
                                int E) {
    int t = blockIdx.x * blockDim.x + threadIdx.x;
    if (t >= E * HEADS) return;
    const int e = t >> 3, h = t & 7;
    const int d = dst[e];
    const float ex = __expf(alpha[t] - amax[(long)d * HEADS + h]);
    alpha[t] = ex;
    atomicAdd(&denom[(long)d * HEADS + h], ex);
}

// ---------------- weighted message scatter-add -----------------------------
__global__ void edge_msg_kernel(const int* __restrict__ src, const int* __restrict__ dst,
                                const float* __restrict__ ew, const float* __restrict__ We,
                                const float* __restrict__ v, const float* __restrict__ alpha,
                                const float* __restrict__ denom, float* __restrict__ o,
                                int E) {
    int t = blockIdx.x * blockDim.x + threadIdx.x;
    if (t >= E * HEADS) return;
    const int e = t >> 3, h = t & 7;
    const int s = src[e], d = dst[e];
    const float a = alpha[t] / (denom[(long)d * HEADS + h] + EPSF);
    const float w = ew[e];
    const float4* vs = (const float4*)(v  + (long)s * HID + h * HC);
    const float4* we = (const float4*)(We + h * HC);
    float4 v0 = vs[0], v1 = vs[1];
    float4 w0 = we[0], w1 = we[1];
    float* od = o + (long)d * HID + h * HC;
    atomicAdd(&od[0], (v0.x + w * w0.x) * a);
    atomicAdd(&od[1], (v0.y + w * w0.y) * a);
    atomicAdd(&od[2], (v0.z + w * w0.z) * a);
    atomicAdd(&od[3], (v0.w + w * w0.w) * a);
    atomicAdd(&od[4], (v1.x + w * w1.x) * a);
    atomicAdd(&od[5], (v1.y + w * w1.y) * a);
    atomicAdd(&od[6], (v1.z + w * w1.z) * a);
    atomicAdd(&od[7], (v1.w + w * w1.w) * a);
}

// ---------------- relu / copy ----------------------------------------------
__global__ void relu_copy_kernel(const float* __restrict__ s, float* __restrict__ d,
                                 int n, int doRelu) {
    int i = blockIdx.x * blockDim.x + threadIdx.x;
    if (i < n) {
        float x = s[i];
        d[i] = doRelu ? fmaxf(x, 0.f) : x;
    }
}

// ---------------- global max pool over graphs ------------------------------
__global__ void pool_max_kernel(const float* __restrict__ hbuf, const int* __restrict__ batch,
                                float* __restrict__ pooled, int N) {
    int t = blockIdx.x * blockDim.x + threadIdx.x;
    if (t >= N * HID) return;
    const int n = t >> 6, j = t & 63;
    atomicMaxF(&pooled[(long)batch[n] * HID + j], hbuf[t]);
}

// ===========================================================================
extern "C" void kernel_launch(void* const* d_in, const int* in_sizes, int n_in,
                              void* d_out, int out_size, void* d_ws, size_t ws_size,
                              hipStream_t stream) {
    const float* x     = (const float*)d_in[0];
    const int*   ei    = (const int*)  d_in[1];
    const float* ew    = (const float*)d_in[2];
    const int*   batch = (const int*)  d_in[3];
    // d_in[4] = num_graphs (device scalar, not needed on host)

    const int N = in_sizes[0] / HID;     // 50000
    const int E = in_sizes[2];           // 800000
    const int G = out_size / OUTC;       // 256

    const int* srcIdx = ei;              // edge_index[0]
    const int* dstIdx = ei + E;          // edge_index[1]

    // params flattened in dict-insertion order:
    // per layer l: Wq,bq,Wk,bk,Wv,bv,Wskip,bskip,We  (9 tensors), then Wlin,blin
    const float* P[29];
    for (int i = 0; i < 29; ++i) P[i] = (const float*)d_in[5 + i];
    const float* Wlin = P[27];
    const float* blin = P[28];

    // ---- workspace layout (floats) ----
    float* ws     = (float*)d_ws;
    float* hbuf   = ws;                       // N*HID  activations
    float* qb     = hbuf   + (long)N * HID;   // N*HID
    float* kb     = qb     + (long)N * HID;   // N*HID
    float* vb     = kb     + (long)N * HID;   // N*HID
    float* ob     = vb     + (long)N * HID;   // N*HID  output accumulator
    float* alpha  = ob     + (long)N * HID;   // E*HEADS
    float* amax   = alpha  + (long)E * HEADS; // N*HEADS
    float* denom  = amax   + (long)N * HEADS; // N*HEADS
    float* pooled = denom  + (long)N * HEADS; // G*HID

    const int EH   = E * HEADS;
    const int NH   = N * HEADS;
    const int NHID = N * HID;
    const int TB   = 256;

    const int gemmBlocks = (N + 31) / 32;     // 32 rows per block (COLT=4)

    for (int l = 0; l < 3; ++l) {
        const float* Wq = P[l*9+0]; const float* bq = P[l*9+1];
        const float* Wk = P[l*9+2]; const float* bk = P[l*9+3];
        const float* Wv = P[l*9+4]; const float* bv = P[l*9+5];
        const float* Wsk= P[l*9+6]; const float* bs = P[l*9+7];
        const float* We = P[l*9+8];
        const float* X  = (l == 0) ? x : hbuf;

        // dense projections via native f32 WMMA (LDS-staged)
        gemm_lds_wmma<4><<<gemmBlocks, 256, 0, stream>>>(X, Wq, bq, qb, N);
        gemm_lds_wmma<4><<<gemmBlocks, 256, 0, stream>>>(X, Wk, bk, kb, N);
        gemm_lds_wmma<4><<<gemmBlocks, 256, 0, stream>>>(X, Wv, bv, vb, N);
        // skip connection seeds the output accumulator
        gemm_lds_wmma<4><<<gemmBlocks, 256, 0, stream>>>(X, Wsk, bs, ob, N);

        fill_kernel<<<(NH + TB - 1) / TB, TB, 0, stream>>>(amax, -INFINITY, NH);
        fill_kernel<<<(NH + TB - 1) / TB, TB, 0, stream>>>(denom, 0.f, NH);

        edge_alpha_kernel<<<(EH + TB - 1) / TB, TB, 0, stream>>>(
            srcIdx, dstIdx, ew, We, qb, kb, alpha, amax, E);
        edge_exp_kernel<<<(EH + TB - 1) / TB, TB, 0, stream>>>(
            dstIdx, amax, alpha, denom, E);
        edge_msg_kernel<<<(EH + TB - 1) / TB, TB, 0, stream>>>(
            srcIdx, dstIdx, ew, We, vb, alpha, denom, ob, E);

        relu_copy_kernel<<<(NHID + TB - 1) / TB, TB, 0, stream>>>(
            ob, hbuf, NHID, (l < 2) ? 1 : 0);
    }

    // global max pool + final linear (HID -> OUTC) straight into d_out
    fill_kernel<<<(G * HID + TB - 1) / TB, TB, 0, stream>>>(pooled, -INFINITY, G * HID);
    pool_max_kernel<<<(NHID + TB - 1) / TB, TB, 0, stream>>>(hbuf, batch, pooled, N);

    // final GEMM: 128 rows per block (COLT=1) -> 2 blocks for G=256
    gemm_lds_wmma<1><<<(G + 127) / 128, 256, 0, stream>>>(pooled, Wlin, blin,
                                                          (float*)d_out, G);
}